// Block_61881888801303
// MI455X (gfx1250) — compile-verified
//
#include <hip/hip_runtime.h>

typedef __attribute__((ext_vector_type(2))) float v2f;
typedef __attribute__((ext_vector_type(4))) float v4f;
typedef __attribute__((ext_vector_type(8))) float v8f;

static constexpr long long HW    = 1024LL * 1024LL;
static constexpr long long NBAT  = 16;
static constexpr long long PLANE = NBAT * 3 * HW;   // 50,331,648 elements per tensor

// ---------------------------------------------------------------------------
// Setup kernel: one wave32 composes the affine chain with V_WMMA_F32_16X16X4_F32.
// d_ws float layout:
//   [ 0.. 8] MA  = W1                 [36..38] bA
//   [ 9..17] MB  = W2*W1              [39..41] bB
//   [18..26] MO  = W5*W4*(W3*MB+W1)   [42..44] bO
//   [27..35] MC  = 2*C0 + 3*MA + 3*MB [45..47] bC
// ---------------------------------------------------------------------------
__global__ void compose_kernel(const float* __restrict__ w1, const float* __restrict__ b1,
                               const float* __restrict__ w2, const float* __restrict__ b2,
                               const float* __restrict__ w3, const float* __restrict__ b3,
                               const float* __restrict__ w4, const float* __restrict__ b4,
                               const float* __restrict__ w5, const float* __restrict__ b5,
                               float* __restrict__ ws)
{
    __shared__ float lds[9 * 256];
    const int lane = threadIdx.x;           // 0..31, EXEC all-ones at WMMA sites
    const int l    = lane & 15;
    const int hiK  = (lane >= 16) ? 2 : 0;  // K offset for A/B operands (16x4 / 4x16)
    const int hiM  = (lane >= 16) ? 8 : 0;  // row offset for C/D (16x16)
    enum { W1o = 0, W2o = 256, W3o = 512, W4o = 768, W5o = 1024,
           BMo = 1280, C0o = 1536, T4o = 1792, O4o = 2048 };

    for (int i = lane; i < 9 * 256; i += 32) lds[i] = 0.0f;   // zero-pad 3x3 -> 16x16
    __syncthreads();
    if (lane < 9) {
        int r = lane / 3, c = lane % 3;
        lds[W1o + r * 16 + c] = w1[lane];
        lds[W2o + r * 16 + c] = w2[lane];
        lds[W3o + r * 16 + c] = w3[lane];
        lds[W4o + r * 16 + c] = w4[lane];
        lds[W5o + r * 16 + c] = w5[lane];
    }
    __syncthreads();

    // D(16x16,f32) = A(16x4) * B(4x16) + C ; operands gathered from LDS per ISA layouts
    auto wmma3 = [&](int aOff, int bOff, int cOff) -> v8f {
        v2f A, B;
        A.x = lds[aOff + l * 16 + hiK + 0];       // A: lane row l, K = hiK / hiK+1
        A.y = lds[aOff + l * 16 + hiK + 1];
        B.x = lds[bOff + (hiK + 0) * 16 + l];     // B: lane col l, K rows hiK / hiK+1
        B.y = lds[bOff + (hiK + 1) * 16 + l];
        v8f C = {};
        if (cOff >= 0) {
            #pragma unroll
            for (int r = 0; r < 8; ++r) C[r] = lds[cOff + (hiM + r) * 16 + l];
        }
        return __builtin_amdgcn_wmma_f32_16x16x4_f32(false, A, false, B, (short)0, C,
                                                     false, false);
    };
    auto storeD = [&](int dOff, v8f D) {
        #pragma unroll
        for (int r = 0; r < 8; ++r) lds[dOff + (hiM + r) * 16 + l] = D[r];
        __syncthreads();
    };

    storeD(BMo, wmma3(W2o, W1o, -1));    // BM = W2*W1
    storeD(C0o, wmma3(W3o, BMo, W1o));   // C0 = W3*BM + W1   (accumulator preloaded)
    storeD(T4o, wmma3(W4o, C0o, -1));    // T4 = W4*C0
    storeD(O4o, wmma3(W5o, T4o, -1));    // O4 = W5*T4

    if (lane == 0) {
        for (int i = 0; i < 9; ++i) {
            int r = i / 3, c = i % 3;
            float A  = lds[W1o + r * 16 + c];
            float BM = lds[BMo + r * 16 + c];
            float C0 = lds[C0o + r * 16 + c];
            float O4 = lds[O4o + r * 16 + c];
            ws[0  + i] = A;
            ws[9  + i] = BM;
            ws[18 + i] = O4;
            ws[27 + i] = 2.0f * C0 + 3.0f * A + 3.0f * BM;
        }
        float bA[3], bB[3], bC0[3], t4[3], bO[3];
        for (int o = 0; o < 3; ++o) bA[o] = b1[o];
        for (int o = 0; o < 3; ++o) { float s = b2[o];
            for (int c = 0; c < 3; ++c) s = fmaf(w2[o*3+c], bA[c], s);  bB[o]  = s; }
        for (int o = 0; o < 3; ++o) { float s = b3[o] + bA[o];
            for (int c = 0; c < 3; ++c) s = fmaf(w3[o*3+c], bB[c], s);  bC0[o] = s; }
        for (int o = 0; o < 3; ++o) { float s = b4[o];
            for (int c = 0; c < 3; ++c) s = fmaf(w4[o*3+c], bC0[c], s); t4[o]  = s; }
        for (int o = 0; o < 3; ++o) { float s = b5[o];
            for (int c = 0; c < 3; ++c) s = fmaf(w5[o*3+c], t4[c], s);  bO[o]  = s; }
        for (int o = 0; o < 3; ++o) {
            ws[36 + o] = bA[o];
            ws[39 + o] = bB[o];
            ws[42 + o] = bO[o];
            ws[45 + o] = 2.0f * bC0[o] + 3.0f * bA[o] + 3.0f * bB[o];
        }
    }
}

// ---------------------------------------------------------------------------
// Hot path: fused streaming kernel. 4 pixels/thread via v4f (b128 global ops),
// non-temporal hints on the 1.4 GB of once-touched traffic. 36 FMA/pixel.
// d_out = out4 | a | bb | c2 | c2 | c2   (each PLANE elements)
// ---------------------------------------------------------------------------
__device__ __forceinline__ v4f fma4(float a, v4f v, v4f acc) {
    acc.x = fmaf(a, v.x, acc.x);
    acc.y = fmaf(a, v.y, acc.y);
    acc.z = fmaf(a, v.z, acc.z);
    acc.w = fmaf(a, v.w, acc.w);
    return acc;
}

__global__ __launch_bounds__(256) void fused_main(const float* __restrict__ x,
                                                  const float* __restrict__ K,
                                                  float* __restrict__ out)
{
    const long long t    = (long long)blockIdx.x * 256 + threadIdx.x;
    const long long p    = t << 2;             // first of this thread's 4 pixels
    const long long n    = p >> 20;            // batch  (HW = 2^20)
    const long long s    = p & (HW - 1);       // spatial offset
    const long long base = n * (3 * HW) + s;

    const v4f X0 = __builtin_nontemporal_load((const v4f*)(x + base));
    const v4f X1 = __builtin_nontemporal_load((const v4f*)(x + base + HW));
    const v4f X2 = __builtin_nontemporal_load((const v4f*)(x + base + 2 * HW));

    float k[48];
    #pragma unroll
    for (int i = 0; i < 48; ++i) k[i] = K[i];   // uniform -> scalar loads

    const int       moff[4] = {18, 0, 9, 27};             // MO, MA, MB, MC
    const int       boff[4] = {42, 36, 39, 45};
    const long long poff[4] = {0, PLANE, 2 * PLANE, 3 * PLANE};

    #pragma unroll
    for (int m = 0; m < 4; ++m) {
        #pragma unroll
        for (int c = 0; c < 3; ++c) {
            const float b = k[boff[m] + c];
            v4f acc = {b, b, b, b};
            acc = fma4(k[moff[m] + c * 3 + 0], X0, acc);
            acc = fma4(k[moff[m] + c * 3 + 1], X1, acc);
            acc = fma4(k[moff[m] + c * 3 + 2], X2, acc);
            __builtin_nontemporal_store(acc, (v4f*)(out + poff[m] + base + c * HW));
            if (m == 3) {   // c2 is replicated 3x in the returned res list
                __builtin_nontemporal_store(acc, (v4f*)(out + 4 * PLANE + base + c * HW));
                __builtin_nontemporal_store(acc, (v4f*)(out + 5 * PLANE + base + c * HW));
            }
        }
    }
}

extern "C" void kernel_launch(void* const* d_in, const int* in_sizes, int n_in,
                              void* d_out, int out_size, void* d_ws, size_t ws_size,
                              hipStream_t stream)
{
    const float* x  = (const float*)d_in[0];
    const float* w1 = (const float*)d_in[1];
    const float* b1 = (const float*)d_in[2];
    const float* w2 = (const float*)d_in[3];
    const float* b2 = (const float*)d_in[4];
    const float* w3 = (const float*)d_in[5];
    const float* b3 = (const float*)d_in[6];
    const float* w4 = (const float*)d_in[7];
    const float* b4 = (const float*)d_in[8];
    const float* w5 = (const float*)d_in[9];
    const float* b5 = (const float*)d_in[10];
    float*       out = (float*)d_out;
    float*       ws  = (float*)d_ws;

    compose_kernel<<<1, 32, 0, stream>>>(w1, b1, w2, b2, w3, b3, w4, b4, w5, b5, ws);

    const long long quads  = NBAT * HW / 4;       // 4,194,304 threads
    const int       blocks = (int)(quads / 256);  // 16,384 blocks
    fused_main<<<blocks, 256, 0, stream>>>(x, ws, out);
}